// MLPGate_62491774157283
// MI455X (gfx1250) — compile-verified
//
#include <hip/hip_runtime.h>
#include <hip/hip_bf16.h>

// ---------------------------------------------------------------------------
// GAT edge scorer: score = h[src]·W_u + h[dst]·W_v + b, min-max normalized.
// Strategy: per-node projection GEMM (100000x128 @ 128x16-padded) via
// V_WMMA_F32_16X16X4_F32, then O(E) gather-add + global min/max + normalize.
// ---------------------------------------------------------------------------

typedef __attribute__((ext_vector_type(2))) float v2f;
typedef __attribute__((ext_vector_type(8))) float v8f;

#define D_FEAT 128

// Monotonic float <-> uint key mapping (order-preserving for atomicMin/Max).
__device__ __forceinline__ unsigned f2k(float f) {
    unsigned x = __float_as_uint(f);
    return (x & 0x80000000u) ? ~x : (x | 0x80000000u);
}
__device__ __forceinline__ float k2f(unsigned k) {
    unsigned x = (k & 0x80000000u) ? (k & 0x7FFFFFFFu) : ~k;
    return __uint_as_float(x);
}
__device__ __forceinline__ unsigned umin2(unsigned a, unsigned b) { return a < b ? a : b; }
__device__ __forceinline__ unsigned umax2(unsigned a, unsigned b) { return a > b ? a : b; }

// ---------------------------------------------------------------------------
// Kernel 0: init min/max keys in workspace.
// ---------------------------------------------------------------------------
__global__ void init_minmax_kernel(unsigned* __restrict__ minmax) {
    minmax[0] = 0xFFFFFFFFu;  // min-key neutral
    minmax[1] = 0x00000000u;  // max-key neutral
}

// ---------------------------------------------------------------------------
// Kernel 1: node projection via WMMA f32 16x16x4.
// Each wave32 handles a tile of 16 node rows. A = h tile (16x4 per step),
// B = [W_u | W_v | 0...] (4x16 per step), C accumulates 16x16 (cols 0,1 used).
//
// A layout (32-bit 16x4): lane L -> row M=L%16; VGPR0 holds K=(L/16)*2,
// VGPR1 holds K=(L/16)*2+1. B is the transposed analog: lane L -> col N=L%16,
// VGPR0 = K=(L/16)*2 row, VGPR1 = next K. C layout: lane L -> col N=L%16,
// VGPR r -> row M = r + 8*(L/16).
// ---------------------------------------------------------------------------
__global__ void node_proj_wmma_kernel(const float* __restrict__ h,
                                      const float* __restrict__ W,
                                      float* __restrict__ u,
                                      float* __restrict__ v,
                                      int n_nodes) {
    const int lane  = threadIdx.x & 31;
    const int wave  = threadIdx.x >> 5;
    const int waves = blockDim.x >> 5;
    const int tile  = blockIdx.x * waves + wave;   // wave-uniform
    const int node0 = tile * 16;
    if (node0 >= n_nodes) return;                  // whole wave exits together

    const int m    = lane & 15;        // A row within tile / B,C column
    const int half = lane >> 4;        // 0: K pair base 0, 1: K pair base 2
    const int kb   = half * 2;

    int row = node0 + m;
    if (row >= n_nodes) row = n_nodes - 1;         // clamp loads; stores guarded
    const float* hrow = h + (size_t)row * D_FEAT;

    // B column select without EXEC divergence (cndmask, not branch):
    // col 0 -> W_u = W[0:128], col 1 -> W_v = W[128:256], cols 2..15 -> 0.
    const float* wsel = W + ((m == 1) ? D_FEAT : 0);
    const float  bmask = (m < 2) ? 1.0f : 0.0f;

    v8f c = {0.f, 0.f, 0.f, 0.f, 0.f, 0.f, 0.f, 0.f};

    #pragma unroll
    for (int k0 = 0; k0 < D_FEAT; k0 += 4) {
        v2f a = *(const v2f*)(hrow + k0 + kb);     // K = k0+kb, k0+kb+1
        v2f bf;
        bf.x = bmask * wsel[k0 + kb];
        bf.y = bmask * wsel[k0 + kb + 1];
        // D = A x B + C   (f32, 16x16x4)
        c = __builtin_amdgcn_wmma_f32_16x16x4_f32(
                /*neg_a=*/false, a, /*neg_b=*/false, bf,
                /*c_mod=*/(short)0, c, /*reuse_a=*/false, /*reuse_b=*/false);
    }

    // Extract column 0 (u) from lanes 0/16, column 1 (v) from lanes 1/17.
    if (m == 0) {
        #pragma unroll
        for (int r = 0; r < 8; ++r) {
            int ro = node0 + r + 8 * half;
            if (ro < n_nodes) u[ro] = c[r];
        }
    } else if (m == 1) {
        #pragma unroll
        for (int r = 0; r < 8; ++r) {
            int ro = node0 + r + 8 * half;
            if (ro < n_nodes) v[ro] = c[r];
        }
    }
}

// ---------------------------------------------------------------------------
// Kernel 2: per-edge score + block min/max reduction into global key pair.
// ---------------------------------------------------------------------------
__global__ void edge_score_kernel(const float* __restrict__ u,
                                  const float* __restrict__ v,
                                  const int* __restrict__ src,
                                  const int* __restrict__ dst,
                                  const float* __restrict__ bias,
                                  float* __restrict__ out,
                                  unsigned* __restrict__ minmax,
                                  int n_edges) {
    __shared__ unsigned smin[256];
    __shared__ unsigned smax[256];

    const int e = blockIdx.x * blockDim.x + threadIdx.x;
    unsigned kmin = 0xFFFFFFFFu;
    unsigned kmax = 0x00000000u;
    if (e < n_edges) {
        float s = u[src[e]] + v[dst[e]] + bias[0];
        out[e] = s;
        unsigned k = f2k(s);
        kmin = k;
        kmax = k;
    }
    smin[threadIdx.x] = kmin;
    smax[threadIdx.x] = kmax;
    __syncthreads();

    for (int off = (int)blockDim.x >> 1; off > 0; off >>= 1) {
        if ((int)threadIdx.x < off) {
            smin[threadIdx.x] = umin2(smin[threadIdx.x], smin[threadIdx.x + off]);
            smax[threadIdx.x] = umax2(smax[threadIdx.x], smax[threadIdx.x + off]);
        }
        __syncthreads();
    }
    if (threadIdx.x == 0) {
        atomicMin(&minmax[0], smin[0]);
        atomicMax(&minmax[1], smax[0]);
    }
}

// ---------------------------------------------------------------------------
// Kernel 3: in-place min-max normalization of d_out.
// ---------------------------------------------------------------------------
__global__ void normalize_kernel(float* __restrict__ out,
                                 const unsigned* __restrict__ minmax,
                                 int n_edges) {
    const int e = blockIdx.x * blockDim.x + threadIdx.x;
    const float lo  = k2f(minmax[0]);
    const float hi  = k2f(minmax[1]);
    const float inv = 1.0f / (hi - lo);
    if (e < n_edges) {
        out[e] = (out[e] - lo) * inv;
    }
}

// ---------------------------------------------------------------------------
// Launch: inputs (h, src, dst, W, b) per setup_inputs() order.
// Workspace layout: [0..255] min/max keys (aligned), then u[N], v[N] floats.
// Raw scores live in d_out and are normalized in place (deterministic).
// ---------------------------------------------------------------------------
extern "C" void kernel_launch(void* const* d_in, const int* in_sizes, int n_in,
                              void* d_out, int out_size, void* d_ws, size_t ws_size,
                              hipStream_t stream) {
    const float* h    = (const float*)d_in[0];
    const int*   src  = (const int*)d_in[1];
    const int*   dst  = (const int*)d_in[2];
    const float* W    = (const float*)d_in[3];
    const float* bias = (const float*)d_in[4];

    const int n_nodes = in_sizes[0] / D_FEAT;
    const int n_edges = in_sizes[1];

    float*    out    = (float*)d_out;
    unsigned* minmax = (unsigned*)d_ws;
    float*    u      = (float*)((char*)d_ws + 256);
    float*    v      = u + n_nodes;

    init_minmax_kernel<<<1, 1, 0, stream>>>(minmax);

    const int tiles       = (n_nodes + 15) / 16;   // 16 nodes per wave
    const int wavesPerBlk = 8;                     // 256 threads
    const int projBlocks  = (tiles + wavesPerBlk - 1) / wavesPerBlk;
    node_proj_wmma_kernel<<<projBlocks, 256, 0, stream>>>(h, W, u, v, n_nodes);

    const int edgeBlocks = (n_edges + 255) / 256;
    edge_score_kernel<<<edgeBlocks, 256, 0, stream>>>(u, v, src, dst, bias,
                                                      out, minmax, n_edges);
    normalize_kernel<<<edgeBlocks, 256, 0, stream>>>(out, minmax, n_edges);
}